// UnFusedGCNLayer_35210141893098
// MI455X (gfx1250) — compile-verified
//
#include <hip/hip_runtime.h>

#define N_NODES 100000
#define N_EDGES 3200000
#define FEAT    128
#define EMBED   128
#define NTILES  (N_NODES / 16)   // 6250 exact
#define WPITCH  132              // LDS row pitch (dwords); ≡4 mod 64 -> conflict-free B-frags

typedef __attribute__((ext_vector_type(2))) float v2f;
typedef __attribute__((ext_vector_type(4))) float v4f;
typedef __attribute__((ext_vector_type(8))) float v8f;
typedef __attribute__((ext_vector_type(4))) unsigned int u32x4;
typedef __attribute__((ext_vector_type(4))) int i32x4;
typedef __attribute__((ext_vector_type(8))) int i32x8;

#ifndef __has_builtin
#define __has_builtin(x) 0
#endif

// ---------------------------------------------------------------------------
// Kernel 1: h = X * W^T via V_WMMA_F32_16X16X4_F32 (full fp32 precision).
// W (64 KB) is DMA'd once per block into LDS by the Tensor Data Mover with
// row padding (pitch 132 dwords) so B-fragment ds_load_b64s are bank-conflict
// free. One wave32 per 16-row stripe of h; 8 column tiles in 8 v8f accums.
// ---------------------------------------------------------------------------
__global__ __launch_bounds__(256)
void gcn_gemm_wmma_f32(const float* __restrict__ X,
                       const float* __restrict__ W,
                       float* __restrict__ H) {
    __shared__ float sW[FEAT * WPITCH];          // 128 rows * 528 B = 67584 B

#if __has_builtin(__builtin_amdgcn_tensor_load_to_lds) && defined(__gfx1250__)
    if (threadIdx.x < 32) {                      // one wave issues the TDM op
        const unsigned long long wa  = (unsigned long long)W;
        const unsigned           lds = (unsigned)(size_t)&sW[0]; // addr[31:0] = LDS offset

        u32x4 g0;
        g0[0] = 1u;                                   // count=1, user descriptor
        g0[1] = lds;                                  // lds_addr (bytes)
        g0[2] = (unsigned)(wa & 0xFFFFFFFFull);       // global_addr[31:0]
        g0[3] = (unsigned)((wa >> 32) & 0x1FFFFFFull) // global_addr[56:32]
              | 0x80000000u;                          // type=2 ("image")

        i32x8 g1;
        g1[0] = (2 << 16)     // data_size = 4 bytes
              | (1 << 20)     // pad_enable
              | (6 << 22)     // pad_interval: every 128 dwords (one 512B row)
              | (3 << 25);    // pad_amount: 4 dwords -> LDS pitch 132 dwords
        g1[1] = (FEAT  & 0xFFFF) << 16;  // tensor_dim0[15:0] = 128
        g1[2] = (EMBED & 0xFFFF) << 16;  // tensor_dim0[31:16]=0 | tensor_dim1[15:0]=128
        g1[3] = (FEAT  & 0xFFFF) << 16;  // tensor_dim1[31:16]=0 | tile_dim0=128
        g1[4] = EMBED;                   // tile_dim1=128 | tile_dim2=0
        g1[5] = FEAT;                    // tensor_dim0_stride[31:0] = 128
        g1[6] = 0;                       // stride0[47:32] | stride1[15:0]
        g1[7] = 0;                       // stride1[47:16]

        i32x4 gz = {0, 0, 0, 0};
#if __clang_major__ >= 23
        i32x8 gz8 = {0, 0, 0, 0, 0, 0, 0, 0};
        __builtin_amdgcn_tensor_load_to_lds(g0, g1, gz, gz, gz8, 0);
#else
        __builtin_amdgcn_tensor_load_to_lds(g0, g1, gz, gz, 0);
#endif
    }
    __builtin_amdgcn_s_wait_tensorcnt(0);
    asm volatile("" ::: "memory");
#else
    // Fallback: cooperative padded copy of W into LDS.
    for (int i = threadIdx.x; i < FEAT * EMBED; i += 256) {
        sW[(i >> 7) * WPITCH + (i & 127)] = W[i];
    }
#endif
    __syncthreads();

    const int lane = threadIdx.x & 31;
    const int wv   = threadIdx.x >> 5;
    const int tile = blockIdx.x * 8 + wv;        // 16-row tile index
    if (tile >= NTILES) return;

    const int row0 = tile * 16;
    const int m    = lane & 15;                  // row (A) / col (B,C) in tile
    const int kh   = (lane >> 4) << 1;           // lane-half K offset: 0 or 2

    const float* xrow = X + (size_t)(row0 + m) * FEAT;

    v8f acc[8] = {};                             // 8 C tiles = one 16x128 stripe

    #pragma unroll 4
    for (int k = 0; k < FEAT; k += 4) {
        // A fragment: X[row0+m][k+kh .. k+kh+1]
        v2f a = *(const v2f*)(xrow + k + kh);
        #pragma unroll
        for (int t = 0; t < 8; ++t) {
            // B fragment: B[f][e] = W[e][f] -> sW row (t*16+m), cols k+kh..+1
            v2f b = *(const v2f*)(&sW[(t * 16 + m) * WPITCH + k + kh]);
            acc[t] = __builtin_amdgcn_wmma_f32_16x16x4_f32(
                false, a, false, b, (short)0, acc[t], false, false);
        }
    }

    // C/D layout: VGPR r, lanes 0-15 -> M=r, lanes 16-31 -> M=r+8; N = lane&15.
    const int mhi = (lane >> 4) * 8;
    #pragma unroll
    for (int t = 0; t < 8; ++t) {
        float* dst = H + (size_t)row0 * EMBED + t * 16 + m;
        #pragma unroll
        for (int r = 0; r < 8; ++r) {
            dst[(size_t)(r + mhi) * EMBED] = acc[t][r];
        }
    }
}

// ---------------------------------------------------------------------------
// Kernel 2: CSR row offsets from the sorted edge_dst array.
// start[n] = lower_bound(edge_dst, n); start[N_NODES] = N_EDGES.
// ---------------------------------------------------------------------------
__global__ __launch_bounds__(256)
void gcn_csr_offsets(const int* __restrict__ edge_dst, int* __restrict__ start) {
    const int n = blockIdx.x * blockDim.x + threadIdx.x;
    if (n > N_NODES) return;
    int lo = 0, hi = N_EDGES;
    while (lo < hi) {
        int mid = (lo + hi) >> 1;
        if (edge_dst[mid] < n) lo = mid + 1; else hi = mid;
    }
    start[n] = lo;
}

// ---------------------------------------------------------------------------
// Kernel 3: out[n] = sum_{e in [start[n],start[n+1])} ew[e] * h[esrc[e]]
// One wave32 per node, each lane owns 4 of 128 channels (float4 gather =
// coalesced 512B per row). Edge streams + out use non-temporal hints so the
// L2 stays dedicated to the h gather working set (51.2 MB < 192 MB L2).
// ---------------------------------------------------------------------------
__global__ __launch_bounds__(256)
void gcn_aggregate(const float* __restrict__ H,
                   const float* __restrict__ ew,
                   const int* __restrict__ esrc,
                   const int* __restrict__ start,
                   float* __restrict__ out) {
    const int lane = threadIdx.x & 31;
    const int wv   = threadIdx.x >> 5;
    const int node = blockIdx.x * 8 + wv;
    if (node >= N_NODES) return;

    const int e0 = start[node];
    const int e1 = start[node + 1];

    v4f acc = {};
    for (int e = e0; e < e1; ++e) {
        const int   s = __builtin_nontemporal_load(&esrc[e]);
        const float w = __builtin_nontemporal_load(&ew[e]);
        v4f v = *(const v4f*)(H + (size_t)s * EMBED + lane * 4);  // RT: L2-resident
        acc += v * w;
    }
    __builtin_nontemporal_store(acc, (v4f*)(out + (size_t)node * EMBED + lane * 4));
}

// ---------------------------------------------------------------------------
extern "C" void kernel_launch(void* const* d_in, const int* in_sizes, int n_in,
                              void* d_out, int out_size, void* d_ws, size_t ws_size,
                              hipStream_t stream) {
    const float* x    = (const float*)d_in[0];   // [N_NODES, FEAT]
    const float* w    = (const float*)d_in[1];   // [EMBED, FEAT]
    const float* ew   = (const float*)d_in[2];   // [N_EDGES]
    const int*   esrc = (const int*)  d_in[3];   // [N_EDGES]
    const int*   edst = (const int*)  d_in[4];   // [N_EDGES], sorted
    float*       out  = (float*)d_out;           // [N_NODES, EMBED]

    // Workspace layout: h (fp32, N*EMBED) then CSR offsets (N+1 ints).
    float* h     = (float*)d_ws;
    int*   start = (int*)((char*)d_ws + (size_t)N_NODES * EMBED * sizeof(float));

    // 1) Dense transform: fp32 WMMA, W staged to LDS by the TDM.
    gcn_gemm_wmma_f32<<<(NTILES + 7) / 8, 256, 0, stream>>>(x, w, h);

    // 2) CSR offsets (same stream keeps ordering).
    gcn_csr_offsets<<<(N_NODES + 1 + 255) / 256, 256, 0, stream>>>(edst, start);

    // 3) Atomic-free segmented reduction, one wave per node.
    gcn_aggregate<<<(N_NODES + 7) / 8, 256, 0, stream>>>(h, ew, esrc, start, out);
}